// RNN_67997922230970
// MI455X (gfx1250) — compile-verified
//
#include <hip/hip_runtime.h>
#include <cstddef>
#include <cstdint>

typedef __attribute__((ext_vector_type(2))) float v2f;
typedef __attribute__((ext_vector_type(8))) float v8f;

#define BATCH   64
#define SEQ     512
#define EMBED   1024
#define HIDDEN  1024
#define VOCAB   32000

#define HBYTES      ((size_t)BATCH * HIDDEN * sizeof(float))        // 256 KB
#define XPROJ_BYTES ((size_t)BATCH * SEQ * HIDDEN * sizeof(float))  // 128 MiB

// ---------------------------------------------------------------------------
// CDNA5 async global->LDS DMA (ASYNCcnt-tracked), 16 bytes per lane.
// ---------------------------------------------------------------------------
static __device__ __forceinline__ unsigned lds_off(const void* p) {
  return (unsigned)(uintptr_t)(const __attribute__((address_space(3))) void*)p;
}

static __device__ __forceinline__ void async_copy16(unsigned lds_byte_off,
                                                    const void* gptr) {
  asm volatile("global_load_async_to_lds_b128 %0, %1, off"
               :
               : "v"(lds_byte_off), "v"((unsigned long long)(uintptr_t)gptr)
               : "memory");
}

static __device__ __forceinline__ void wait_async0() {
  asm volatile("s_wait_asynccnt 0x0" ::: "memory");
}

// ---------------------------------------------------------------------------
// One 16x16 f32 tile, K=1024 reduction with V_WMMA_F32_16X16X4_F32.
// 4 interleaved accumulator chains (depth 64 each) to hide WMMA latency.
// Per-lane operand layout (ISA 7.12.2, 32-bit A 16x4 / B 4x16 / C 16x16):
//   m = lane&15, half = lane>>4
//   A frag = { A[m][k+2*half], A[m][k+2*half+1] }
//   B frag = { B[k+2*half][n], B[k+2*half+1][n] }
//   C vgpr j = C[j + 8*half][n]
// arow points at row base + 2*half; bcol at B + 2*half*ldb + n.
// ---------------------------------------------------------------------------
static __device__ __forceinline__ void tile_mma_k1024(
    const float* __restrict__ arow, const float* __restrict__ bcol,
    int ldb, v8f acc[4])
{
  for (int kk = 0; kk < 1024; kk += 16) {
#pragma unroll
    for (int q = 0; q < 4; ++q) {
      const int k = kk + 4 * q;
      v2f a; a.x = arow[k];                 a.y = arow[k + 1];
      v2f b; b.x = bcol[(size_t)k * ldb];   b.y = bcol[(size_t)(k + 1) * ldb];
      acc[q] = __builtin_amdgcn_wmma_f32_16x16x4_f32(
          false, a, false, b, (short)0, acc[q], false, false);
    }
  }
}

// Same, but B tile lives in LDS as [K][16] (row stride 16 floats).
// blds points at &Bl[(2*half)*16 + n].
static __device__ __forceinline__ void tile_mma_k1024_ldsB(
    const float* __restrict__ arow, const float* __restrict__ blds,
    v8f acc[4])
{
  for (int kk = 0; kk < 1024; kk += 16) {
#pragma unroll
    for (int q = 0; q < 4; ++q) {
      const int k = kk + 4 * q;
      v2f a; a.x = arow[k];        a.y = arow[k + 1];
      v2f b; b.x = blds[k * 16];   b.y = blds[(k + 1) * 16];
      acc[q] = __builtin_amdgcn_wmma_f32_16x16x4_f32(
          false, a, false, b, (short)0, acc[q], false, false);
    }
  }
}

// ---------------------------------------------------------------------------
// Phase 1: xproj[b*SEQ+t][h] = embedding[x[b*SEQ+t]] @ W_xh + b_h
// M=32768, N=1024, K=1024.  Block = 256 thr (8 waves) computes 64x64.
// Double-buffered LDS fed by async global->LDS DMA: issue chunk k+1,
// compute chunk k, s_wait_asynccnt 0, one barrier per chunk.
// Strides chosen so both wave-halves hit disjoint LDS banks.
// ---------------------------------------------------------------------------
#define A_STRIDE 20
#define B_STRIDE 80

__global__ __launch_bounds__(256) void embed_xproj_kernel(
    const int*   __restrict__ x,
    const float* __restrict__ emb,
    const float* __restrict__ Wxh,
    const float* __restrict__ bh,
    float*       __restrict__ xproj)
{
  __shared__ float As[2][64 * A_STRIDE];   // 2 x 5120 B
  __shared__ float Bs[2][16 * B_STRIDE];   // 2 x 5120 B

  const int tid = threadIdx.x;
  const int m0  = blockIdx.x * 64;      // 512 M blocks
  const int n0  = blockIdx.y * 64;      // 16  N blocks

  // A-DMA assignment: 4 threads per gathered row, 16 B each
  const int arow = tid >> 2;            // 0..63
  const int aseg = (tid & 3) * 4;       // 0,4,8,12
  const float* agp = emb + (size_t)x[m0 + arow] * EMBED + aseg;

  // B-DMA assignment: 16 threads per k-row, 16 B each
  const int brow = tid >> 4;            // 0..15
  const int bcol = (tid & 15) * 4;
  const float* bgp = Wxh + (size_t)brow * HIDDEN + n0 + bcol;

  const unsigned a_lds[2] = { lds_off(&As[0][arow * A_STRIDE + aseg]),
                              lds_off(&As[1][arow * A_STRIDE + aseg]) };
  const unsigned b_lds[2] = { lds_off(&Bs[0][brow * B_STRIDE + bcol]),
                              lds_off(&Bs[1][brow * B_STRIDE + bcol]) };

  // compute assignment: wave w -> M tile (w>>1), N tiles (w&1)*2 and +1
  const int lane  = tid & 31;
  const int wave  = tid >> 5;
  const int tm    = wave >> 1;
  const int tn0   = (wave & 1) * 2;
  const int half  = lane >> 4;
  const int mlane = lane & 15;

  v8f acc0 = {}; v8f acc1 = {};

  // prologue: DMA chunk 0 into buffer 0
  async_copy16(a_lds[0], agp);
  async_copy16(b_lds[0], bgp);
  wait_async0();
  __syncthreads();

  int p = 0;
  for (int k0 = 0; k0 < EMBED; k0 += 16) {
    if (k0 + 16 < EMBED) {                       // DMA next chunk
      async_copy16(a_lds[p ^ 1], agp + (k0 + 16));
      async_copy16(b_lds[p ^ 1], bgp + (size_t)(k0 + 16) * HIDDEN);
    }
#pragma unroll
    for (int kk = 0; kk < 16; kk += 4) {         // compute current chunk
      const float* ap = &As[p][(tm * 16 + mlane) * A_STRIDE + kk + 2 * half];
      v2f a; a.x = ap[0]; a.y = ap[1];
      const float* bp = &Bs[p][(kk + 2 * half) * B_STRIDE + mlane];
      v2f b0; b0.x = bp[tn0 * 16];      b0.y = bp[B_STRIDE + tn0 * 16];
      v2f b1; b1.x = bp[tn0 * 16 + 16]; b1.y = bp[B_STRIDE + tn0 * 16 + 16];
      acc0 = __builtin_amdgcn_wmma_f32_16x16x4_f32(false, a, false, b0,
                                                   (short)0, acc0, false, false);
      acc1 = __builtin_amdgcn_wmma_f32_16x16x4_f32(false, a, false, b1,
                                                   (short)0, acc1, false, false);
    }
    wait_async0();        // my wave's DMA for next chunk landed
    __syncthreads();      // everyone's DMA landed; everyone done reading buf p
    p ^= 1;
  }

  const int gn0 = n0 + tn0 * 16 + mlane;
  const float bias0 = bh[gn0], bias1 = bh[gn0 + 16];
#pragma unroll
  for (int j = 0; j < 8; ++j) {
    const size_t gm = (size_t)(m0 + tm * 16 + j + 8 * half);
    xproj[gm * HIDDEN + gn0]      = acc0[j] + bias0;
    xproj[gm * HIDDEN + gn0 + 16] = acc1[j] + bias1;
  }
}

// ---------------------------------------------------------------------------
// Phase 2: persistent recurrence.  256 blocks x 1 wave; block = one 16x16
// tile of h (4 row strips x 64 col tiles).  The block's loop-invariant W_hh
// column tile (1024x16 = 64 KB) is DMA'd into LDS once and reused for all
// 512 steps.  Double-buffered h + monotonic device-scope barrier between
// steps.  If xproj==nullptr (workspace too small), fuse the input
// projection (gathered embedding @ W_xh) per step.
// ---------------------------------------------------------------------------
__global__ __launch_bounds__(32) void rnn_steps_kernel(
    const float* __restrict__ xproj,    // may be null -> fused mode
    const int*   __restrict__ xtok,
    const float* __restrict__ emb,
    const float* __restrict__ Wxh,
    const float* __restrict__ bh,
    const float* __restrict__ Whh,
    float*       __restrict__ hbuf0,    // holds h0 = 0 at entry
    float*       __restrict__ hbuf1,
    unsigned int* __restrict__ bar)
{
  __shared__ float Bl[HIDDEN * 16];     // 64 KB: W_hh[:, tn*16 .. +16)

  const int lane  = threadIdx.x;
  const int tm    = blockIdx.x >> 6;    // 0..3   (16-row strip of batch)
  const int tn    = blockIdx.x & 63;    // 0..63  (16-col tile of hidden)
  const int half  = lane >> 4;
  const int mlane = lane & 15;
  const int gn    = tn * 16 + mlane;
  const int brow  = tm * 16 + mlane;    // batch row feeding this lane's A frag

  // ---- preload W_hh column tile into LDS (once) via async DMA ----
  {
    const float* wsrc = Whh + tn * 16;
    for (int i = 0; i < 128; ++i) {
      const int idx = i * 32 + lane;    // 0..4095 16-byte segments
      const int k   = idx >> 2;
      const int seg = (idx & 3) << 2;
      async_copy16(lds_off(&Bl[k * 16 + seg]),
                   wsrc + (size_t)k * HIDDEN + seg);
    }
    wait_async0();
    __syncthreads();
  }

  const float* blds    = Bl + (2 * half) * 16 + mlane;
  const float* bcol_xh = Wxh + (size_t)(2 * half) * HIDDEN + gn;

  for (int t = 0; t < SEQ; ++t) {
    const float* hin  = (t & 1) ? hbuf1 : hbuf0;
    float*       hout = (t & 1) ? hbuf0 : hbuf1;

    v8f acc[4] = {};
    tile_mma_k1024_ldsB(hin + (size_t)brow * HIDDEN + 2 * half, blds, acc);
    if (xproj == nullptr) {
      const size_t tok = (size_t)xtok[brow * SEQ + t];
      tile_mma_k1024(emb + tok * EMBED + 2 * half, bcol_xh, HIDDEN, acc);
    }
    v8f s = acc[0] + acc[1] + acc[2] + acc[3];

#pragma unroll
    for (int j = 0; j < 8; ++j) {
      const int row = tm * 16 + j + 8 * half;
      float add = xproj ? xproj[((size_t)row * SEQ + t) * HIDDEN + gn]
                        : bh[gn];
      hout[(size_t)row * HIDDEN + gn] = tanhf(s[j] + add);
    }

    // ---- grid barrier (monotonic counter, release/acquire) ----
    __threadfence();
    if (lane == 0)
      __hip_atomic_fetch_add(bar, 1u, __ATOMIC_ACQ_REL, __HIP_MEMORY_SCOPE_AGENT);
    const unsigned int target = 256u * (unsigned int)(t + 1);
    while (__hip_atomic_load(bar, __ATOMIC_ACQUIRE, __HIP_MEMORY_SCOPE_AGENT)
           < target) {
      __builtin_amdgcn_s_sleep(2);
    }
  }
}

// ---------------------------------------------------------------------------
// Phase 3: out = h_final @ W_hy + b_y.  2000 blocks x 4 waves; wave w = M
// tile w, block = one 16-wide N tile -> W_hy streamed once from HBM
// (131 MB / 23.3 TB/s ~ 5.6 us floor).
// ---------------------------------------------------------------------------
__global__ __launch_bounds__(128) void out_proj_kernel(
    const float* __restrict__ h,
    const float* __restrict__ Why,
    const float* __restrict__ by,
    float*       __restrict__ out)
{
  const int lane  = threadIdx.x & 31;
  const int wave  = threadIdx.x >> 5;   // 0..3 -> M tile
  const int half  = lane >> 4;
  const int mlane = lane & 15;
  const int gn    = blockIdx.x * 16 + mlane;

  v8f acc[4] = {};
  tile_mma_k1024(h + (size_t)(wave * 16 + mlane) * HIDDEN + 2 * half,
                 Why + (size_t)(2 * half) * VOCAB + gn, VOCAB, acc);
  v8f s = acc[0] + acc[1] + acc[2] + acc[3];

  const float bias = by[gn];
#pragma unroll
  for (int j = 0; j < 8; ++j) {
    const int row = wave * 16 + j + 8 * half;
    out[(size_t)row * VOCAB + gn] = s[j] + bias;
  }
}

// ---------------------------------------------------------------------------
extern "C" void kernel_launch(void* const* d_in, const int* in_sizes, int n_in,
                              void* d_out, int out_size, void* d_ws,
                              size_t ws_size, hipStream_t stream)
{
  const int*   x   = (const int*)  d_in[0];  // (64, 512) token ids
  const float* emb = (const float*)d_in[1];  // (32000, 1024)
  const float* Wxh = (const float*)d_in[2];  // (1024, 1024)
  const float* Whh = (const float*)d_in[3];  // (1024, 1024)
  const float* bh  = (const float*)d_in[4];  // (1024,)
  const float* Why = (const float*)d_in[5];  // (1024, 32000)
  const float* by  = (const float*)d_in[6];  // (32000,)
  float*       out = (float*)d_out;          // (64, 32000)

  char* ws = (char*)d_ws;
  unsigned int* counter = (unsigned int*)ws;            // 4 KB reserved
  float* h0 = (float*)(ws + 4096);
  float* h1 = (float*)(ws + 4096 + HBYTES);
  const size_t need = 4096 + 2 * HBYTES + XPROJ_BYTES;
  float* xproj = (ws_size >= need) ? (float*)(ws + 4096 + 2 * HBYTES) : nullptr;

  // zero the barrier counter and h0 (h starts at zero) every call
  hipMemsetAsync(d_ws, 0, 4096 + HBYTES, stream);

  if (xproj) {
    embed_xproj_kernel<<<dim3(512, 16), 256, 0, stream>>>(x, emb, Wxh, bh,
                                                          xproj);
  }
  rnn_steps_kernel<<<256, 32, 0, stream>>>(xproj, x, emb, Wxh, bh, Whh,
                                           h0, h1, counter);
  out_proj_kernel<<<VOCAB / 16, 128, 0, stream>>>(h0, Why, by, out);
}